// NEATNCLBaseLearner_73684458930294
// MI455X (gfx1250) — compile-verified
//
#include <hip/hip_runtime.h>

// ---------------------------------------------------------------------------
// NEAT-style sparse layers densified into per-layer GEMMs on the CDNA5
// matrix pipe (V_WMMA_F32_16X16X4_F32, full fp32 precision).
//
// values buffer V: [8192 x 1344] f32 in d_ws (fits in 192MB L2).
// dense transposed weights Wt (per layer): Wt[s][i] = sum of w[s,d] over d
// with src_idx[s,d]==i.  agg[b,s] = sum_i V[b,i]*Wt[s][i].
// ---------------------------------------------------------------------------

typedef __attribute__((ext_vector_type(2))) float v2f;
typedef __attribute__((ext_vector_type(8))) float v8f;

#define BATCH   8192
#define N_IN    256
#define N_EVAL  1088
#define DEG     32
#define VWIDTH  1344   // 256 + 256+256+256+256+64
#define N_OUT   64

// dense transposed weight region offsets (floats): layer l has sz_l rows of
// K_l = 256*(l+1) floats.  sizes: 65536,131072,196608,262144,81920 -> 737280
#define WT_TOTAL 737280

// ---------------------------------------------------------------------------
// Build dense transposed weights.  One thread per neuron: zero its row, then
// serially scatter-add its 32 (index, weight) pairs (duplicates accumulate,
// no atomics needed).
// ---------------------------------------------------------------------------
__global__ void build_weights(const float* __restrict__ w,
                              const int*   __restrict__ src,
                              float*       __restrict__ wt)
{
    int t = blockIdx.x * blockDim.x + threadIdx.x;
    if (t >= N_EVAL) return;

    int l = t >> 8; if (l > 4) l = 4;      // layer id 0..4
    const int row_start = l << 8;          // 0,256,512,768,1024
    const int K = (l + 1) << 8;            // input width 256..1280
    const int wbase[5] = {0, 65536, 196608, 393216, 655360};

    const int s = t - row_start;
    float* row = wt + wbase[l] + (size_t)s * K;
    for (int i = 0; i < K; ++i) row[i] = 0.0f;

    const float* wr = w + (size_t)t * DEG;
    const int*   sr = src + (size_t)t * DEG;
#pragma unroll
    for (int d = 0; d < DEG; ++d) row[sr[d]] += wr[d];
}

// ---------------------------------------------------------------------------
// Seed V[:,0:256] = x.  float4 copies.
// ---------------------------------------------------------------------------
__global__ void copy_x(const float* __restrict__ x, float* __restrict__ V)
{
    int t  = blockIdx.x * blockDim.x + threadIdx.x;   // BATCH * N_IN/4 threads
    int b  = t >> 6;                                  // N_IN/4 == 64
    int c4 = t & 63;
    ((float4*)(V + (size_t)b * VWIDTH))[c4] =
        ((const float4*)(x + (size_t)b * N_IN))[c4];
}

// ---------------------------------------------------------------------------
// One layer:  agg = V[:, :K] @ Wt^T ; out = relu(bias + resp*agg)
// Each wave owns a 32x32 output tile: 2x2 grid of 16x16 fp32 WMMA accums,
// K-loop in steps of 4 via V_WMMA_F32_16X16X4_F32.
//
// fp32 A (16x4) lane layout:  lanes 0-15: M=lane, VGPR0/1 = K=0/1
//                             lanes 16-31: M=lane-16, VGPR0/1 = K=2/3
// fp32 B (4x16) lane layout:  lanes 0-15: N=lane, VGPR0/1 = K=0/1
//                             lanes 16-31: N=lane-16, VGPR0/1 = K=2/3
// -> both are contiguous float2 loads (A row-major in V, B rows of Wt).
// fp32 C/D (16x16): VGPR r: lanes 0-15 M=r, lanes 16-31 M=r+8; N = lane&15.
// ---------------------------------------------------------------------------
template<int K, int SZ>
__global__ void layer_gemm(const float* __restrict__ Wt,
                           const float* __restrict__ bias,
                           const float* __restrict__ resp,
                           float* V,           // read cols [0,K), write [K,K+SZ)
                           float* out,         // non-null only on last layer
                           int row_start)
{
    const int lane = threadIdx.x & 31;
    const int wave = threadIdx.x >> 5;
    const int tile = blockIdx.x * (blockDim.x >> 5) + wave;

    const int ntn = SZ / 32;
    const int tm  = (tile / ntn) * 32;
    const int tn  = (tile % ntn) * 32;

    const int hl  = lane & 15;          // M (for A) / N (for B) within 16
    const int kph = (lane >> 4) * 2;    // K phase: 0 or 2

    const float* a0 = V  + (size_t)(tm + hl)      * VWIDTH + kph;
    const float* a1 = V  + (size_t)(tm + 16 + hl) * VWIDTH + kph;
    const float* b0 = Wt + (size_t)(tn + hl)      * K      + kph;
    const float* b1 = Wt + (size_t)(tn + 16 + hl) * K      + kph;

    v8f c00 = {}, c01 = {}, c10 = {}, c11 = {};

#pragma unroll 2
    for (int k0 = 0; k0 < K; k0 += 4) {
        v2f A0 = *(const v2f*)(a0 + k0);
        v2f A1 = *(const v2f*)(a1 + k0);
        v2f B0 = *(const v2f*)(b0 + k0);
        v2f B1 = *(const v2f*)(b1 + k0);
        c00 = __builtin_amdgcn_wmma_f32_16x16x4_f32(false, A0, false, B0,
                                                    (short)0, c00, false, false);
        c01 = __builtin_amdgcn_wmma_f32_16x16x4_f32(false, A0, false, B1,
                                                    (short)0, c01, false, false);
        c10 = __builtin_amdgcn_wmma_f32_16x16x4_f32(false, A1, false, B0,
                                                    (short)0, c10, false, false);
        c11 = __builtin_amdgcn_wmma_f32_16x16x4_f32(false, A1, false, B1,
                                                    (short)0, c11, false, false);
    }

    // Epilogue: relu(bias + resp*agg), write next column block of V
    const int nlo  = lane & 15;
    const int mofs = (lane >> 4) * 8;

    const int n0 = tn + nlo;        // N for c00 / c10
    const int n1 = tn + 16 + nlo;   // N for c01 / c11
    const float bi0 = bias[row_start + n0], re0 = resp[row_start + n0];
    const float bi1 = bias[row_start + n1], re1 = resp[row_start + n1];

#pragma unroll
    for (int r = 0; r < 8; ++r) {
        const int m0 = tm + mofs + r;        // M for c00 / c01
        const int m1 = tm + 16 + mofs + r;   // M for c10 / c11
        float o00 = bi0 + re0 * c00[r]; o00 = o00 > 0.0f ? o00 : 0.0f;
        float o01 = bi1 + re1 * c01[r]; o01 = o01 > 0.0f ? o01 : 0.0f;
        float o10 = bi0 + re0 * c10[r]; o10 = o10 > 0.0f ? o10 : 0.0f;
        float o11 = bi1 + re1 * c11[r]; o11 = o11 > 0.0f ? o11 : 0.0f;
        V[(size_t)m0 * VWIDTH + K + n0] = o00;
        V[(size_t)m0 * VWIDTH + K + n1] = o01;
        V[(size_t)m1 * VWIDTH + K + n0] = o10;
        V[(size_t)m1 * VWIDTH + K + n1] = o11;
        if (out) {   // last layer: SZ == N_OUT, columns map 1:1
            out[(size_t)m0 * N_OUT + n0] = o00;
            out[(size_t)m0 * N_OUT + n1] = o01;
            out[(size_t)m1 * N_OUT + n0] = o10;
            out[(size_t)m1 * N_OUT + n1] = o11;
        }
    }
}

// ---------------------------------------------------------------------------
extern "C" void kernel_launch(void* const* d_in, const int* in_sizes, int n_in,
                              void* d_out, int out_size, void* d_ws, size_t ws_size,
                              hipStream_t stream)
{
    const float* x    = (const float*)d_in[0];   // [8192,256]
    const float* w    = (const float*)d_in[1];   // [1088,32]
    const float* bias = (const float*)d_in[2];   // [1088]
    const float* resp = (const float*)d_in[3];   // [1088]
    const int*   src  = (const int*)  d_in[4];   // [1088,32]

    float* wt  = (float*)d_ws;                   // 737280 floats (~2.95 MB)
    float* V   = wt + WT_TOTAL;                  // 8192*1344 floats (~42 MB)
    float* out = (float*)d_out;                  // [8192,64]

    build_weights<<<(N_EVAL + 255) / 256, 256, 0, stream>>>(w, src, wt);
    copy_x<<<(BATCH * (N_IN / 4)) / 256, 256, 0, stream>>>(x, V);

    // 8 waves/block, each wave one 32x32 tile; grids divide exactly:
    // SZ=256: 256 M-tiles * 8 N-tiles = 2048 tiles -> 256 blocks
    // SZ=64 : 256 M-tiles * 2 N-tiles =  512 tiles ->  64 blocks
    layer_gemm< 256,256><<<256, 256, 0, stream>>>(wt + 0,      bias, resp, V, nullptr,    0);
    layer_gemm< 512,256><<<256, 256, 0, stream>>>(wt + 65536,  bias, resp, V, nullptr,  256);
    layer_gemm< 768,256><<<256, 256, 0, stream>>>(wt + 196608, bias, resp, V, nullptr,  512);
    layer_gemm<1024,256><<<256, 256, 0, stream>>>(wt + 393216, bias, resp, V, nullptr,  768);
    layer_gemm<1280, 64><<< 64, 256, 0, stream>>>(wt + 655360, bias, resp, V, out,     1024);
}